// Encoder_20942260535802
// MI455X (gfx1250) — compile-verified
//
#include <hip/hip_runtime.h>
#include <hip/hip_bf16.h>

// ---------------------------------------------------------------------------
// RETAIN-style encoder for MI455X (gfx1250, wave32, WMMA + TDM).
//   VOCAB=4880, EDIM=128, HDIM=128, NVISITS=8192
// Pipeline:
//   1) VEP[s] = H^T @ E (K-split x4)   WMMA f32 16x16x4, E staged via TDM->LDS
//   1b) VE   = sum_s VEP[s]            float4 reduction
//   2) XP  = VE @ W_ih^T + b_ih        WMMA f32 16x16x4
//   3) HID = GRU scan (serial, W_hh rows in VGPRs)
//   4) out = softmax(HID @ attn_w) @ HID
// ---------------------------------------------------------------------------

#define VOCAB   4880
#define EDIM    128
#define HDIM    128
#define NVISITS 8192
#define GDIM    (3 * HDIM)      // 384

#define KSPLIT  4
#define KGROUP  (VOCAB / KSPLIT)    // 1220
#define KCHUNK  20                  // rows of E staged per TDM transfer
#define NCHUNK  (KGROUP / KCHUNK)   // 61

typedef float v2f __attribute__((ext_vector_type(2)));
typedef float v8f __attribute__((ext_vector_type(8)));
typedef unsigned int v4ui __attribute__((ext_vector_type(4)));
typedef int v4i __attribute__((ext_vector_type(4)));
typedef int v8i __attribute__((ext_vector_type(8)));

// WMMA f32 16x16x4 lane mapping (ISA 7.12.2):
//  A (16x4 MxK):  lane L -> M = L&15 ; VGPR0 = K=kb, VGPR1 = K=kb+1, kb = 2*(L>>4)
//  B (4x16 KxN):  lane L -> N = L&15 ; VGPR0 = K=kb, VGPR1 = K=kb+1 (assumed symmetric)
//  C/D (16x16):   VGPR r -> M = r + 8*(L>>4) ; N = L&15

// ---------------------------------------------------------------------------
// TDM 2-D tile load: Global -> LDS. Descriptor packed per CDNA5 ISA
// sections 8.3 (group0) / 8.4 (group1); groups 2-3 zero (<=2D tensor).
// amdgpu-toolchain (clang-23) 6-arg builtin form.
// ---------------------------------------------------------------------------
__device__ __forceinline__ void tdm_load_2d(unsigned int lds_byte_off,
                                            const float* gsrc,
                                            int rows,        // tile_dim1
                                            int cols,        // tile_dim0
                                            int row_stride)  // tensor_dim0_stride (elems)
{
    const unsigned long long ga = (unsigned long long)(uintptr_t)gsrc;
    v4ui g0;
    g0.x = 0x1u;                                   // count=1, user descriptor
    g0.y = lds_byte_off;                           // lds_addr [63:32]
    g0.z = (unsigned int)(ga & 0xFFFFFFFFu);       // global_addr [95:64]
    g0.w = (unsigned int)((ga >> 32) & 0x01FFFFFFu)// global_addr [120:96]
         | (2u << 30);                             // type=2 ("image") [127:126]
    v8i g1;
    g1[0] = (int)(2u << 16);                       // data_size=2 (4 bytes); mask=0
    g1[1] = (int)((unsigned)cols << 16);           // tensor_dim0[15:0] @ bit 48+
    g1[2] = (int)((unsigned)rows << 16);           // tensor_dim0 hi=0 | tensor_dim1[15:0]
    g1[3] = (int)((unsigned)cols << 16);           // tensor_dim1 hi=0 | tile_dim0
    g1[4] = (int)((unsigned)rows & 0xFFFFu);       // tile_dim1 | tile_dim2=0
    g1[5] = row_stride;                            // tensor_dim0_stride[31:0]
    g1[6] = 0;                                     // stride hi | tensor_dim1_stride lo
    g1[7] = 0;
    const v4i gz4 = {0, 0, 0, 0};
    const v8i gz8 = {0, 0, 0, 0, 0, 0, 0, 0};
    __builtin_amdgcn_tensor_load_to_lds(g0, g1, gz4, gz4, gz8, 0);
}

__device__ __forceinline__ unsigned int lds_offset_of(const void* p) {
    // addrspace(3) offset == low 32 bits of the flat shared-aperture address
    return (unsigned int)(uintptr_t)p;
}

// ---------------------------------------------------------------------------
// GEMM1 (K-split partial): VEP[kg][t,e] = sum_{v in kgroup} H[v,t] * E[v,e]
// Grid (64, 4) x 256 threads (8 waves). Wave = one 16-row M tile, 8 N tiles.
// H streamed once -> non-temporal loads. E tile double-buffered via TDM->LDS.
// ---------------------------------------------------------------------------
__global__ __launch_bounds__(256) void k_gemm1(const float* __restrict__ H,
                                               const float* __restrict__ E,
                                               float* __restrict__ VEP) {
    __shared__ __align__(16) float ebuf[2][KCHUNK * EDIM];   // 2 x 10 KB

    const int lane = threadIdx.x & 31;
    const int wave = threadIdx.x >> 5;                 // 0..7
    const int m0   = blockIdx.x * 128 + wave * 16;     // visit (t) base
    const int kg0  = blockIdx.y * KGROUP;              // K-group base
    float* vout    = VEP + (size_t)blockIdx.y * NVISITS * EDIM;

    const int mrow = lane & 15;
    const int kb   = (lane >> 4) * 2;                  // 0 or 2

    v8f acc[8];
    v8f zero = {};
#pragma unroll
    for (int i = 0; i < 8; ++i) acc[i] = zero;

    if (wave == 0)
        tdm_load_2d(lds_offset_of(&ebuf[0][0]), E + (size_t)kg0 * EDIM,
                    KCHUNK, EDIM, EDIM);

    for (int c = 0; c < NCHUNK; ++c) {
        __syncthreads();   // everyone done reading ebuf[(c+1)&1] from iter c-1
        if (wave == 0) {
            if (c + 1 < NCHUNK) {
                tdm_load_2d(lds_offset_of(&ebuf[(c + 1) & 1][0]),
                            E + (size_t)(kg0 + (c + 1) * KCHUNK) * EDIM,
                            KCHUNK, EDIM, EDIM);
                __builtin_amdgcn_s_wait_tensorcnt((short)1);  // chunk c done (in-order)
            } else {
                __builtin_amdgcn_s_wait_tensorcnt((short)0);  // last chunk done
            }
        }
        __syncthreads();   // publish ebuf[c&1]
        const float* eb  = &ebuf[c & 1][0];
        const int kbase  = kg0 + c * KCHUNK;
#pragma unroll
        for (int kk = 0; kk < KCHUNK; kk += 4) {
            v2f a;
            a.x = __builtin_nontemporal_load(
                      &H[(size_t)(kbase + kk + kb)     * NVISITS + m0 + mrow]);
            a.y = __builtin_nontemporal_load(
                      &H[(size_t)(kbase + kk + kb + 1) * NVISITS + m0 + mrow]);
#pragma unroll
            for (int nt = 0; nt < 8; ++nt) {
                const int ncol = nt * 16 + (lane & 15);
                v2f b;
                b.x = eb[(kk + kb)     * EDIM + ncol];
                b.y = eb[(kk + kb + 1) * EDIM + ncol];
                acc[nt] = __builtin_amdgcn_wmma_f32_16x16x4_f32(
                    false, a, false, b, (short)0, acc[nt], false, false);
            }
        }
    }

    const int crow = (lane >> 4) * 8;   // 0 or 8
    const int ccol = lane & 15;
#pragma unroll
    for (int nt = 0; nt < 8; ++nt)
#pragma unroll
        for (int r = 0; r < 8; ++r)
            vout[(size_t)(m0 + crow + r) * EDIM + nt * 16 + ccol] = acc[nt][r];
}

// ---------------------------------------------------------------------------
// Reduce the 4 K-split partials: VE = VEP[0] + VEP[1] + VEP[2] + VEP[3]
// ---------------------------------------------------------------------------
__global__ __launch_bounds__(256) void k_reduce_ve(const float* __restrict__ VEP,
                                                   float* __restrict__ VE) {
    const size_t i = ((size_t)blockIdx.x * 256 + threadIdx.x) * 4;
    const size_t stride = (size_t)NVISITS * EDIM;
    float4 s = *(const float4*)(VEP + i);
#pragma unroll
    for (int p = 1; p < KSPLIT; ++p) {
        const float4 v = *(const float4*)(VEP + p * stride + i);
        s.x += v.x; s.y += v.y; s.z += v.z; s.w += v.w;
    }
    *(float4*)(VE + i) = s;
}

// ---------------------------------------------------------------------------
// GEMM2: XP[t,g] = sum_k VE[t,k] * W_ih[g,k] + b_ih[g].  M=8192 K=128 N=384.
// Grid (64, 3) x 256 threads; blockIdx.y picks a 128-wide slice of N.
// ---------------------------------------------------------------------------
__global__ __launch_bounds__(256) void k_gemm2(const float* __restrict__ VE,
                                               const float* __restrict__ W_ih,
                                               const float* __restrict__ b_ih,
                                               float* __restrict__ XP) {
    const int lane = threadIdx.x & 31;
    const int wave = threadIdx.x >> 5;
    const int m0   = blockIdx.x * 128 + wave * 16;
    const int n0   = blockIdx.y * 128;
    const int mrow = lane & 15;
    const int kb   = (lane >> 4) * 2;

    v8f acc[8];
    v8f zero = {};
#pragma unroll
    for (int i = 0; i < 8; ++i) acc[i] = zero;

#pragma unroll 4
    for (int k0 = 0; k0 < EDIM; k0 += 4) {
        v2f a;
        a.x = VE[(size_t)(m0 + mrow) * EDIM + k0 + kb];
        a.y = VE[(size_t)(m0 + mrow) * EDIM + k0 + kb + 1];
#pragma unroll
        for (int nt = 0; nt < 8; ++nt) {
            const int ncol = n0 + nt * 16 + (lane & 15);
            v2f b;
            b.x = W_ih[(size_t)ncol * EDIM + k0 + kb];     // W_ih^T[k,n] = W_ih[n,k]
            b.y = W_ih[(size_t)ncol * EDIM + k0 + kb + 1];
            acc[nt] = __builtin_amdgcn_wmma_f32_16x16x4_f32(
                false, a, false, b, (short)0, acc[nt], false, false);
        }
    }

    const int crow = (lane >> 4) * 8;
    const int ccol = lane & 15;
#pragma unroll
    for (int nt = 0; nt < 8; ++nt) {
        const int g = n0 + nt * 16 + ccol;
        const float bias = b_ih[g];
#pragma unroll
        for (int r = 0; r < 8; ++r)
            XP[(size_t)(m0 + crow + r) * GDIM + g] = acc[nt][r] + bias;
    }
}

// ---------------------------------------------------------------------------
// GRU scan: 1 block, 384 threads. Thread j owns W_hh row j in 128 VGPRs.
// h broadcast via LDS (ds_load_b128). Two barriers per step, 8192 steps.
// Gate order matches reference: [r | z | n].
// ---------------------------------------------------------------------------
__global__ __launch_bounds__(384) void k_gru(const float* __restrict__ XP,
                                             const float* __restrict__ W_hh,
                                             const float* __restrict__ b_hh,
                                             float* __restrict__ HID) {
    __shared__ __align__(16) float h[HDIM];
    __shared__ float gh[GDIM];
    const int j = threadIdx.x;                         // 0..383

    float w[HDIM];
#pragma unroll
    for (int k = 0; k < HDIM; ++k) w[k] = W_hh[(size_t)j * HDIM + k];
    const float bj = b_hh[j];

    if (j < HDIM) h[j] = 0.0f;
    __syncthreads();

    for (int t = 0; t < NVISITS; ++t) {
        float acc = bj;
#pragma unroll
        for (int k = 0; k < HDIM; k += 4) {
            const float4 hv = *(const float4*)&h[k];   // LDS broadcast read
            acc += w[k] * hv.x + w[k + 1] * hv.y + w[k + 2] * hv.z + w[k + 3] * hv.w;
        }
        gh[j] = acc;
        __syncthreads();

        if (j < HDIM) {
            const float xr = XP[(size_t)t * GDIM + j];
            const float xz = XP[(size_t)t * GDIM + HDIM + j];
            const float xn = XP[(size_t)t * GDIM + 2 * HDIM + j];
            const float r  = 1.0f / (1.0f + __expf(-(xr + gh[j])));
            const float z  = 1.0f / (1.0f + __expf(-(xz + gh[HDIM + j])));
            const float n  = tanhf(xn + r * gh[2 * HDIM + j]);
            const float hn = (1.0f - z) * n + z * h[j];
            h[j] = hn;
            HID[(size_t)t * HDIM + j] = hn;
        }
        __syncthreads();
    }
}

// ---------------------------------------------------------------------------
// Attention scores: S[t] = dot(HID[t,:], attn_w). One thread per row (float4).
// ---------------------------------------------------------------------------
__global__ __launch_bounds__(256) void k_scores(const float* __restrict__ HID,
                                                const float* __restrict__ attn_w,
                                                float* __restrict__ S) {
    const int t = blockIdx.x * 256 + threadIdx.x;
    const float4* hv = (const float4*)(HID + (size_t)t * HDIM);
    const float4* av = (const float4*)attn_w;
    float acc = 0.0f;
#pragma unroll
    for (int k = 0; k < HDIM / 4; ++k) {
        const float4 hh = hv[k];
        const float4 aa = av[k];
        acc += hh.x * aa.x + hh.y * aa.y + hh.z * aa.z + hh.w * aa.w;
    }
    S[t] = acc;
}

// ---------------------------------------------------------------------------
// Softmax over 8192 scores + weighted sum. Single block (deterministic
// fixed-order tree reductions), 256 threads; lanes 0..127 accumulate out[e].
// ---------------------------------------------------------------------------
__global__ __launch_bounds__(256) void k_softmax_out(const float* __restrict__ S,
                                                     const float* __restrict__ HID,
                                                     float* __restrict__ alpha,
                                                     float* __restrict__ out) {
    __shared__ float red[256];
    const int tid = threadIdx.x;

    float m = -3.402823466e38f;
    for (int t = tid; t < NVISITS; t += 256) m = fmaxf(m, S[t]);
    red[tid] = m;
    __syncthreads();
    for (int s = 128; s > 0; s >>= 1) {
        if (tid < s) red[tid] = fmaxf(red[tid], red[tid + s]);
        __syncthreads();
    }
    const float mx = red[0];
    __syncthreads();

    float sum = 0.0f;
    for (int t = tid; t < NVISITS; t += 256) {
        const float e = __expf(S[t] - mx);
        alpha[t] = e;
        sum += e;
    }
    red[tid] = sum;
    __syncthreads();
    for (int s = 128; s > 0; s >>= 1) {
        if (tid < s) red[tid] += red[tid + s];
        __syncthreads();
    }
    const float inv = 1.0f / red[0];
    __syncthreads();

    if (tid < HDIM) {
        float acc = 0.0f;
        for (int t = 0; t < NVISITS; ++t)
            acc += alpha[t] * HID[(size_t)t * HDIM + tid];   // coalesced across lanes
        out[tid] = acc * inv;
    }
}

// ---------------------------------------------------------------------------
extern "C" void kernel_launch(void* const* d_in, const int* in_sizes, int n_in,
                              void* d_out, int out_size, void* d_ws, size_t ws_size,
                              hipStream_t stream) {
    const float* H      = (const float*)d_in[0];   // (4880, 8192)
    const float* E      = (const float*)d_in[1];   // (4880, 128)
    const float* W_ih   = (const float*)d_in[2];   // (384, 128)
    const float* W_hh   = (const float*)d_in[3];   // (384, 128)
    const float* b_ih   = (const float*)d_in[4];   // (384,)
    const float* b_hh   = (const float*)d_in[5];   // (384,)
    const float* attn_w = (const float*)d_in[6];   // (128,)
    float* out = (float*)d_out;                    // (128,)

    float* ws    = (float*)d_ws;
    float* VEP   = ws;                                       // 4 * 8192*128
    float* VE    = VEP + (size_t)KSPLIT * NVISITS * EDIM;    // 8192*128
    float* XP    = VE  + (size_t)NVISITS * EDIM;             // 8192*384
    float* HID   = XP  + (size_t)NVISITS * GDIM;             // 8192*128
    float* S     = HID + (size_t)NVISITS * HDIM;             // 8192
    float* ALPHA = S + NVISITS;                              // 8192

    k_gemm1<<<dim3(NVISITS / 128, KSPLIT), dim3(256), 0, stream>>>(H, E, VEP);
    k_reduce_ve<<<dim3(NVISITS * EDIM / 4 / 256), dim3(256), 0, stream>>>(VEP, VE);
    k_gemm2<<<dim3(NVISITS / 128, 3), dim3(256), 0, stream>>>(VE, W_ih, b_ih, XP);
    k_gru<<<dim3(1), dim3(GDIM), 0, stream>>>(XP, W_hh, b_hh, HID);
    k_scores<<<dim3(NVISITS / 256), dim3(256), 0, stream>>>(HID, attn_w, S);
    k_softmax_out<<<dim3(1), dim3(256), 0, stream>>>(S, HID, ALPHA, out);
}